// HyperbolicGraphConvolution_21131239097068
// MI455X (gfx1250) — compile-verified
//
#include <hip/hip_runtime.h>
#include <math.h>

typedef __attribute__((ext_vector_type(2))) float v2f;
typedef __attribute__((ext_vector_type(8))) float v8f;

#define NROW 1024
#define DDIM 64
#define MINC 1e-15f
#define EPSC 1e-7f

__device__ __forceinline__ float artanh_c(float y) {
    y = fminf(fmaxf(y, -1.0f + EPSC), 1.0f - EPSC);
    return 0.5f * (log1pf(y) - log1pf(-y));
}

// ---------------------------------------------------------------------------
// K1: per-row transform.  mx = x @ weight^T via V_WMMA_F32_16X16X4_F32,
// then mobius_matvec scaling, mobius_add(bias), logmap0 -> u,v dots, |h|^2.
// Also emits H transposed (Ht) so the K=1024 GEMM gets contiguous B loads.
// Block = 256 thr (8 waves) handles 32 rows.
// ---------------------------------------------------------------------------
__global__ __launch_bounds__(256) void k_row_transform(
    const float* __restrict__ x, const float* __restrict__ weight,
    const float* __restrict__ bias, const float* __restrict__ attw,
    float* __restrict__ H, float* __restrict__ Ht, float* __restrict__ x2a,
    float* __restrict__ ua, float* __restrict__ va)
{
    __shared__ float smem[32][DDIM];
    const int tid  = threadIdx.x;
    const int wv   = tid >> 5, lane = tid & 31;
    const int mrow = wv >> 2, ncol = wv & 3;
    const int rowBase = blockIdx.x * 32;
    const int lm = lane & 15, lk = lane >> 4;
    const int m  = rowBase + mrow * 16 + lm;   // A row
    const int e  = ncol * 16 + lm;             // B column (weight output idx)
    const int koff = 2 * lk;

    v8f acc0 = {}, acc1 = {};
    for (int t = 0; t < 8; ++t) {
        int k0 = 8 * t + koff;
        v2f a0 = *(const v2f*)(x + m * DDIM + k0);
        v2f b0 = *(const v2f*)(weight + e * DDIM + k0);
        acc0 = __builtin_amdgcn_wmma_f32_16x16x4_f32(
            false, a0, false, b0, (short)0, acc0, false, false);
        v2f a1 = *(const v2f*)(x + m * DDIM + k0 + 4);
        v2f b1 = *(const v2f*)(weight + e * DDIM + k0 + 4);
        acc1 = __builtin_amdgcn_wmma_f32_16x16x4_f32(
            false, a1, false, b1, (short)0, acc1, false, false);
    }
    v8f acc = acc0 + acc1;
#pragma unroll
    for (int v = 0; v < 8; ++v)
        smem[mrow * 16 + v + 8 * lk][ncol * 16 + lm] = acc[v];
    __syncthreads();

    if (tid < 32) {
        const int r = rowBase + tid;
        float xn2 = 0.f;
        for (int d = 0; d < DDIM; ++d) { float xv = x[r * DDIM + d]; xn2 += xv * xv; }
        float xn = fmaxf(sqrtf(xn2), MINC);
        float mxn2 = 0.f;
        for (int d = 0; d < DDIM; ++d) { float mv = smem[tid][d]; mxn2 += mv * mv; }
        float mxn = fmaxf(sqrtf(mxn2), MINC);
        float sc  = tanhf((mxn / xn) * artanh_c(xn)) / mxn;
        if (mxn2 == 0.f) sc = 0.f;                      // all-zero mx row -> zeros
        for (int d = 0; d < DDIM; ++d) smem[tid][d] *= sc;   // res in place
        // hyp_bias = expmap0(bias)
        float bn2 = 0.f;
        for (int d = 0; d < DDIM; ++d) bn2 += bias[d] * bias[d];
        float bn  = fmaxf(sqrtf(bn2), MINC);
        float bsc = tanhf(bn) / bn;
        // mobius_add(res, hyp_bias)
        float rx2 = 0.f, ry2 = 0.f, rxy = 0.f;
        for (int d = 0; d < DDIM; ++d) {
            float a0 = smem[tid][d], b0 = bsc * bias[d];
            rx2 += a0 * a0; ry2 += b0 * b0; rxy += a0 * b0;
        }
        float cx  = 1.f + 2.f * rxy + ry2, cy = 1.f - rx2;
        float den = fmaxf(1.f + 2.f * rxy + rx2 * ry2, MINC);
        float hx2 = 0.f;
        for (int d = 0; d < DDIM; ++d) {
            float hv = (cx * smem[tid][d] + cy * bsc * bias[d]) / den;
            smem[tid][d] = hv; hx2 += hv * hv;
        }
        // logmap0(h) -> attention dots u,v
        float hn = fmaxf(sqrtf(hx2), MINC);
        float ls = artanh_c(hn) / hn;
        float uu = 0.f, vv = 0.f;
        for (int d = 0; d < DDIM; ++d) {
            float hv = smem[tid][d];
            float ht = ls * hv;
            uu += ht * attw[d]; vv += ht * attw[DDIM + d];
            H[r * DDIM + d]  = hv;
            Ht[d * NROW + r] = hv;
        }
        x2a[r] = hx2; ua[r] = uu; va[r] = vv;
    }
}

// ---------------------------------------------------------------------------
// K2 (fused Gram + pairwise): compute xy = <h_i,h_j> tile via f32 WMMA and
// apply the pairwise hyperbolic scalar map directly on the accumulator:
//   att = sigmoid(u_i + v_j + b) * adj ;  sub = a*h_i + b*h_j
//   g = max(1-x2,MIN) * artanh(sn)/sn
//   W_mat[i][j] = att*g*b ;  s-contribution att*g*a reduced deterministically
// Block covers 32(i) x 64(j); grid (32, 16).  Partial row-sums go to
// sp[i][blockIdx.y] so the final reduction order is fixed (graph-replay safe).
// ---------------------------------------------------------------------------
__global__ __launch_bounds__(256) void k_gram_pairwise(
    const float* __restrict__ H, const float* __restrict__ adj,
    const float* __restrict__ x2a, const float* __restrict__ ua,
    const float* __restrict__ va, const float* __restrict__ attb,
    float* __restrict__ Wm, float* __restrict__ sp)
{
    __shared__ float scon[32][DDIM];   // per-(i_local, j_local) s-contribution
    const int tid = threadIdx.x, wv = tid >> 5, lane = tid & 31;
    const int isub = wv >> 2, jsub = wv & 3;
    const int lm = lane & 15, lk = lane >> 4, koff = 2 * lk;
    const int iBase = blockIdx.x * 32;
    const int iA = iBase + isub * 16 + lm;
    const int j  = blockIdx.y * 64 + jsub * 16 + lm;

    v8f acc0 = {}, acc1 = {};
    for (int t = 0; t < 8; ++t) {
        int k0 = 8 * t + koff;
        v2f a0 = *(const v2f*)(H + iA * DDIM + k0);
        v2f b0 = *(const v2f*)(H + j * DDIM + k0);
        acc0 = __builtin_amdgcn_wmma_f32_16x16x4_f32(
            false, a0, false, b0, (short)0, acc0, false, false);
        v2f a1 = *(const v2f*)(H + iA * DDIM + k0 + 4);
        v2f b1 = *(const v2f*)(H + j * DDIM + k0 + 4);
        acc1 = __builtin_amdgcn_wmma_f32_16x16x4_f32(
            false, a1, false, b1, (short)0, acc1, false, false);
    }
    v8f acc = acc0 + acc1;

    const float ab = attb[0];
    const float y2 = x2a[j];
    const float vj = va[j];
    const int   jl = jsub * 16 + lm;
#pragma unroll
    for (int v = 0; v < 8; ++v) {
        const int il = isub * 16 + v + 8 * lk;   // 0..31 within block
        const int i  = iBase + il;
        const float x2 = x2a[i];
        const float ui = ua[i];
        const float xy = acc[v];
        const float adjv = adj[(size_t)i * NROW + j];
        const float att  = adjv / (1.f + expf(-(ui + vj + ab)));
        const float den  = fmaxf(1.f - 2.f * xy + x2 * y2, MINC);
        const float a    = -(1.f - 2.f * xy + y2) / den;
        const float b    = (1.f - x2) / den;
        const float sn2  = a * a * x2 + 2.f * a * b * xy + b * b * y2;
        const float sn   = fmaxf(sqrtf(fmaxf(sn2, 0.f)), MINC);
        const float g    = fmaxf(1.f - x2, MINC) * artanh_c(sn) / sn;
        const float wg   = att * g;
        Wm[(size_t)i * NROW + j] = wg * b;
        scon[il][jl] = wg * a;
    }
    __syncthreads();
    if (tid < 32) {                 // fixed-order (deterministic) row reduce
        float s = 0.f;
        for (int q = 0; q < DDIM; ++q) s += scon[tid][q];
        sp[(size_t)(iBase + tid) * 16 + blockIdx.y] = s;
    }
}

// ---------------------------------------------------------------------------
// K3: S = W_mat @ H  (1024 x 64, K = 1024) via f32 WMMA with Ht for
// contiguous B loads and 4 independent accumulator chains for ILP.
// ---------------------------------------------------------------------------
__global__ __launch_bounds__(256) void k_support(
    const float* __restrict__ Wm, const float* __restrict__ Ht,
    float* __restrict__ S)
{
    const int tid = threadIdx.x, wv = tid >> 5, lane = tid & 31;
    const int mrow = wv >> 2, ncol = wv & 3;
    const int lm = lane & 15, lk = lane >> 4, koff = 2 * lk;
    const int m = blockIdx.x * 32 + mrow * 16 + lm;
    const int n = ncol * 16 + lm;
    v8f c0 = {}, c1 = {}, c2 = {}, c3 = {};
    for (int k = 0; k < NROW; k += 16) {
        v2f a0 = *(const v2f*)(Wm + (size_t)m * NROW + k + koff);
        v2f b0 = *(const v2f*)(Ht + (size_t)n * NROW + k + koff);
        c0 = __builtin_amdgcn_wmma_f32_16x16x4_f32(
            false, a0, false, b0, (short)0, c0, false, false);
        v2f a1 = *(const v2f*)(Wm + (size_t)m * NROW + k + 4 + koff);
        v2f b1 = *(const v2f*)(Ht + (size_t)n * NROW + k + 4 + koff);
        c1 = __builtin_amdgcn_wmma_f32_16x16x4_f32(
            false, a1, false, b1, (short)0, c1, false, false);
        v2f a2 = *(const v2f*)(Wm + (size_t)m * NROW + k + 8 + koff);
        v2f b2 = *(const v2f*)(Ht + (size_t)n * NROW + k + 8 + koff);
        c2 = __builtin_amdgcn_wmma_f32_16x16x4_f32(
            false, a2, false, b2, (short)0, c2, false, false);
        v2f a3 = *(const v2f*)(Wm + (size_t)m * NROW + k + 12 + koff);
        v2f b3 = *(const v2f*)(Ht + (size_t)n * NROW + k + 12 + koff);
        c3 = __builtin_amdgcn_wmma_f32_16x16x4_f32(
            false, a3, false, b3, (short)0, c3, false, false);
    }
    v8f acc = (c0 + c1) + (c2 + c3);
#pragma unroll
    for (int v = 0; v < 8; ++v) {
        int io = blockIdx.x * 32 + mrow * 16 + v + 8 * lk;
        S[io * DDIM + ncol * 16 + lm] = acc[v];
    }
}

// ---------------------------------------------------------------------------
// K4: per-row tail: s_i = sum_b sp[i][b] (fixed order);
// support = s_i*h_i + S_i; out = expmap(support @ h);
// then HypAct: expmap0(relu(logmap0(out))).  One thread per row.
// ---------------------------------------------------------------------------
__global__ __launch_bounds__(256) void k_finalize(
    const float* __restrict__ H, const float* __restrict__ S,
    const float* __restrict__ x2a, const float* __restrict__ sp,
    float* __restrict__ out)
{
    const int r = blockIdx.x * 256 + threadIdx.x;
    if (r >= NROW) return;
    float h[DDIM], w[DDIM];
    float si = 0.f;
    for (int b = 0; b < 16; ++b) si += sp[(size_t)r * 16 + b];
    const float x2 = x2a[r];
    float un2 = 0.f;
    for (int d = 0; d < DDIM; ++d) {
        h[d] = H[r * DDIM + d];
        float u = si * h[d] + S[r * DDIM + d];
        w[d] = u; un2 += u * u;
    }
    float un  = fmaxf(sqrtf(un2), MINC);
    float lam = 2.f / fmaxf(1.f - x2, MINC);
    float sc  = tanhf(0.5f * lam * un) / un;   // second = sc * u
    float y2 = 0.f, xy = 0.f;
    for (int d = 0; d < DDIM; ++d) { w[d] *= sc; y2 += w[d] * w[d]; xy += h[d] * w[d]; }
    float cx  = 1.f + 2.f * xy + y2, cy = 1.f - x2;
    float den = fmaxf(1.f + 2.f * xy + x2 * y2, MINC);
    float on2 = 0.f;
    for (int d = 0; d < DDIM; ++d) {
        float o = (cx * h[d] + cy * w[d]) / den;
        w[d] = o; on2 += o * o;
    }
    float on = fmaxf(sqrtf(on2), MINC);
    float ls = artanh_c(on) / on;
    float tn2 = 0.f;
    for (int d = 0; d < DDIM; ++d) {
        float t = fmaxf(ls * w[d], 0.f);       // relu(logmap0)
        w[d] = t; tn2 += t * t;
    }
    float tn = fmaxf(sqrtf(tn2), MINC);
    float fs = tanhf(tn) / tn;                 // expmap0
    for (int d = 0; d < DDIM; ++d) out[r * DDIM + d] = fs * w[d];
}

// ---------------------------------------------------------------------------
extern "C" void kernel_launch(void* const* d_in, const int* in_sizes, int n_in,
                              void* d_out, int out_size, void* d_ws, size_t ws_size,
                              hipStream_t stream)
{
    (void)in_sizes; (void)n_in; (void)out_size; (void)ws_size;
    const float* x      = (const float*)d_in[0];
    const float* adj    = (const float*)d_in[1];
    const float* weight = (const float*)d_in[2];
    const float* bias   = (const float*)d_in[3];
    const float* attw   = (const float*)d_in[4];
    const float* attb   = (const float*)d_in[5];

    float* ws  = (float*)d_ws;
    float* H   = ws;                                    // N*D
    float* Ht  = H   + (size_t)NROW * DDIM;             // D*N
    float* Wm  = Ht  + (size_t)NROW * DDIM;             // N*N
    float* S   = Wm  + (size_t)NROW * NROW;             // N*D
    float* x2a = S   + (size_t)NROW * DDIM;             // N
    float* ua  = x2a + NROW;                            // N
    float* va  = ua  + NROW;                            // N
    float* sp  = va  + NROW;                            // N*16 partial row sums
    float* out = (float*)d_out;

    k_row_transform<<<32, 256, 0, stream>>>(x, weight, bias, attw, H, Ht, x2a, ua, va);
    k_gram_pairwise<<<dim3(32, 16), 256, 0, stream>>>(H, adj, x2a, ua, va, attb, Wm, sp);
    k_support<<<32, 256, 0, stream>>>(Wm, Ht, S);
    k_finalize<<<4, 256, 0, stream>>>(H, S, x2a, sp, out);
}